// CalibrationLoss_58248346468830
// MI455X (gfx1250) — compile-verified
//
#include <hip/hip_runtime.h>
#include <cstdint>

#define NBINS 10
#define CCOLS 128                 // logits columns
#define ROWS_PER_BATCH 8
#define WAVES_PER_BLOCK 8
#define BLOCK (WAVES_PER_BLOCK * 32)
#define GRID 2048

#if __has_builtin(__builtin_amdgcn_global_load_async_to_lds_b128) && \
    __has_builtin(__builtin_amdgcn_s_wait_asynccnt)
#define HAVE_ASYNC 1
#else
#define HAVE_ASYNC 0
#endif

typedef int v4i __attribute__((ext_vector_type(4)));
typedef __attribute__((address_space(1))) v4i* gv4ip;  // pointer into global
typedef __attribute__((address_space(3))) v4i* lv4ip;  // pointer into LDS

__device__ __forceinline__ gv4ip to_global_v4(const void* p) {
  // Global aperture: flat VA == global VA.
  return (gv4ip)(unsigned long long)(uintptr_t)p;
}
__device__ __forceinline__ lv4ip to_lds_v4(const void* p) {
  // LDS aperture: flat addr low 32 bits == LDS byte offset.
  return (lv4ip)(uint32_t)(uintptr_t)p;
}

// ---------------------------------------------------------------------------
// Pass 1: per-row softmax-confidence + argmax, binned into 3*NBINS float
// accumulators (counts, sum_conf, sum_acc) in global workspace.
// One wave per row; rows staged through LDS with CDNA5 async loads.
// ---------------------------------------------------------------------------
__global__ void __launch_bounds__(BLOCK)
ece_pass1(const float* __restrict__ logits, const int* __restrict__ targets,
          float* __restrict__ bins, int N) {
  __shared__ float stage[WAVES_PER_BLOCK][ROWS_PER_BATCH][CCOLS];
  __shared__ float s_cnt[NBINS], s_conf[NBINS], s_acc[NBINS];

  const int lane = threadIdx.x & 31;
  const int wid = threadIdx.x >> 5;
  const int gwave = blockIdx.x * WAVES_PER_BLOCK + wid;
  const int nwave = GRID * WAVES_PER_BLOCK;

  if (threadIdx.x < NBINS) {
    s_cnt[threadIdx.x] = 0.0f;
    s_conf[threadIdx.x] = 0.0f;
    s_acc[threadIdx.x] = 0.0f;
  }
  __syncthreads();

  float* myStage = &stage[wid][0][0];

  for (int base = gwave * ROWS_PER_BATCH; base < N;
       base += nwave * ROWS_PER_BATCH) {
    const int nrows = min(ROWS_PER_BATCH, N - base);

#if HAVE_ASYNC
    // Stream this wave's batch into its LDS slice with the async engine.
    for (int r = 0; r < nrows; ++r) {
      const float* gsrc = logits + (size_t)(base + r) * CCOLS + lane * 4;
      float* ldst = myStage + r * CCOLS + lane * 4;
      __builtin_amdgcn_global_load_async_to_lds_b128(to_global_v4(gsrc),
                                                     to_lds_v4(ldst), 0, 0);
    }
    __builtin_amdgcn_s_wait_asynccnt(0);
#else
    for (int r = 0; r < nrows; ++r) {
      ((float4*)(myStage + r * CCOLS))[lane] =
          ((const float4*)(logits + (size_t)(base + r) * CCOLS))[lane];
    }
#endif

    for (int r = 0; r < nrows; ++r) {
      const float4 v = ((const float4*)(myStage + r * CCOLS))[lane];

      // Lane-local max + argmax (first occurrence wins on ties).
      float m = v.x;
      int mi = lane * 4;
      if (v.y > m) { m = v.y; mi = lane * 4 + 1; }
      if (v.z > m) { m = v.z; mi = lane * 4 + 2; }
      if (v.w > m) { m = v.w; mi = lane * 4 + 3; }

      // Wave32 butterfly reduction for (max, argmax), lowest index on ties.
      for (int off = 16; off > 0; off >>= 1) {
        const float om = __shfl_xor(m, off, 32);
        const int oi = __shfl_xor(mi, off, 32);
        if (om > m || (om == m && oi < mi)) { m = om; mi = oi; }
      }

      // Sum of exp(x - max): confidence = 1 / sum.
      float s = expf(v.x - m) + expf(v.y - m) + expf(v.z - m) + expf(v.w - m);
      for (int off = 16; off > 0; off >>= 1) s += __shfl_xor(s, off, 32);

      if (lane == 0) {
        const float conf = 1.0f / s;
        int b = (int)ceilf(conf * (float)NBINS) - 1;
        b = (b < 0) ? 0 : ((b > NBINS - 1) ? NBINS - 1 : b);
        const float acc = (mi == targets[base + r]) ? 1.0f : 0.0f;
        atomicAdd(&s_cnt[b], 1.0f);
        atomicAdd(&s_conf[b], conf);
        atomicAdd(&s_acc[b], acc);
      }
    }
  }

  __syncthreads();
  if (threadIdx.x < NBINS) {
    atomicAdd(&bins[threadIdx.x], s_cnt[threadIdx.x]);
    atomicAdd(&bins[NBINS + threadIdx.x], s_conf[threadIdx.x]);
    atomicAdd(&bins[2 * NBINS + threadIdx.x], s_acc[threadIdx.x]);
  }
}

// Zero the 30-float workspace (done every launch: deterministic, graph-safe).
__global__ void ece_init(float* __restrict__ bins) {
  if (threadIdx.x < 3 * NBINS) bins[threadIdx.x] = 0.0f;
}

// Final scalar combine over 10 bins.
__global__ void ece_final(const float* __restrict__ bins,
                          float* __restrict__ out, float invN) {
  if (threadIdx.x == 0) {
    float e = 0.0f;
    for (int b = 0; b < NBINS; ++b) {
      const float c = bins[b];
      if (c > 0.0f) {
        const float avg_conf = bins[NBINS + b] / c;
        const float avg_acc = bins[2 * NBINS + b] / c;
        e += fabsf(avg_conf - avg_acc) * c * invN;
      }
    }
    *out = e;
  }
}

extern "C" void kernel_launch(void* const* d_in, const int* in_sizes, int n_in,
                              void* d_out, int out_size, void* d_ws,
                              size_t ws_size, hipStream_t stream) {
  const float* logits = (const float*)d_in[0];
  const int* targets = (const int*)d_in[1];
  const int N = in_sizes[1];  // targets element count == number of rows
  float* bins = (float*)d_ws; // 3*NBINS floats of scratch
  float* out = (float*)d_out;

  hipLaunchKernelGGL(ece_init, dim3(1), dim3(32), 0, stream, bins);
  hipLaunchKernelGGL(ece_pass1, dim3(GRID), dim3(BLOCK), 0, stream, logits,
                     targets, bins, N);
  hipLaunchKernelGGL(ece_final, dim3(1), dim3(32), 0, stream, bins, out,
                     1.0f / (float)N);
}